// RobustMambaDiagLTI_53317724012877
// MI455X (gfx1250) — compile-verified
//
#include <hip/hip_runtime.h>

// ---------------- fixed problem dims (reference setup_inputs) ----------------
#define BB   4
#define TT   2048
#define DD   1024
#define NN   1024
#define BT   8192            // B*T
#define SIXN 6144            // 6*N  (u_tilde | delta | sign | b | c | d)
#define NC   16              // scan chunks
#define CL   128             // chunk length (NC*CL == TT)

typedef __attribute__((ext_vector_type(2))) float v2f;
typedef __attribute__((ext_vector_type(8))) float v8f;

// ---------------------------------------------------------------------------
// Power iteration: largest singular value of n x n row-major W.
// One workgroup of n (=1024) threads.  out[0] = 1 / max(sigma, 1).
// ---------------------------------------------------------------------------
__global__ __launch_bounds__(1024)
void spectral_inv_scale_kernel(const float* __restrict__ W, int n,
                               float* __restrict__ out_inv) {
  __shared__ float sv[1024];
  __shared__ float sw[1024];
  __shared__ float red[1024];
  const int t = threadIdx.x;
  sv[t] = 1.0f;
  __syncthreads();
  float sigma = 0.0f;
  for (int it = 0; it < 32; ++it) {
    // w = W v   (row dot; W row cached in L2 after first sweep)
    const float* row = W + (size_t)t * n;
    float acc = 0.0f;
    for (int j = 0; j < n; ++j) acc += row[j] * sv[j];
    sw[t]  = acc;
    red[t] = acc * acc;
    __syncthreads();
    for (int s = 512; s > 0; s >>= 1) {
      if (t < s) red[t] += red[t + s];
      __syncthreads();
    }
    const float winv = rsqrtf(red[0] + 1e-30f);
    __syncthreads();
    // v' = W^T (w/||w||)  (coalesced column access)
    float acc2 = 0.0f;
    for (int i = 0; i < n; ++i) acc2 += W[(size_t)i * n + t] * (sw[i] * winv);
    red[t] = acc2 * acc2;
    __syncthreads();
    for (int s = 512; s > 0; s >>= 1) {
      if (t < s) red[t] += red[t + s];
      __syncthreads();
    }
    sigma = sqrtf(red[0]);          // ||W^T w_hat|| -> largest singular value
    const float vinv = (sigma > 0.0f) ? (1.0f / sigma) : 0.0f;
    __syncthreads();
    sv[t] = acc2 * vinv;
    __syncthreads();
  }
  if (t == 0) out_inv[0] = 1.0f / fmaxf(sigma, 1.0f);
}

// ---------------------------------------------------------------------------
// fp32 WMMA GEMM:  C[row, colOff+col] = (A @ B^T) * (*scale) + bias[col]
//   A: M x K row-major, B: Nf x K row-major, C leading dim = ldc.
// 128x128 tile / workgroup; 8 waves, each 64x32 (4x2 grid of 16x16 WMMA).
// ---------------------------------------------------------------------------
#define LPITCH 20   // LDS row pitch (floats): 16-row fragment reads hit 16 distinct banks

__global__ __launch_bounds__(256)
void gemm_wmma_f32_kernel(const float* __restrict__ A, const float* __restrict__ B,
                          float* __restrict__ C, int M, int Nf, int K,
                          int ldc, int colOff,
                          const float* __restrict__ scale_ptr,
                          const float* __restrict__ bias) {
  __shared__ float As[128 * LPITCH];
  __shared__ float Bs[128 * LPITCH];

  const int tid  = threadIdx.x;
  const int lane = tid & 31;
  const int wave = tid >> 5;
  const int bm = blockIdx.y * 128;
  const int bn = blockIdx.x * 128;
  const int waveM = (wave >> 2) * 64;   // 0 / 64
  const int waveN = (wave & 3) * 32;    // 0,32,64,96
  const int ml = lane & 15;
  const int kp = (lane < 16) ? 0 : 2;

  v8f acc[4][2];
  const v8f vzero = {0.f, 0.f, 0.f, 0.f, 0.f, 0.f, 0.f, 0.f};
  for (int i = 0; i < 4; ++i)
    for (int j = 0; j < 2; ++j) acc[i][j] = vzero;

  for (int k0 = 0; k0 < K; k0 += 16) {
    // stage 128x16 tiles of A and B (float4 per thread x2)
    for (int r = 0; r < 2; ++r) {
      const int idx  = tid * 2 + r;        // 0..511 float4 slots = 2048 floats
      const int row  = idx >> 2;
      const int col4 = (idx & 3) * 4;
      float4 av = *(const float4*)(A + (size_t)(bm + row) * K + k0 + col4);
      float* da = &As[row * LPITCH + col4];
      da[0] = av.x; da[1] = av.y; da[2] = av.z; da[3] = av.w;
      float4 bv = *(const float4*)(B + (size_t)(bn + row) * K + k0 + col4);
      float* db = &Bs[row * LPITCH + col4];
      db[0] = bv.x; db[1] = bv.y; db[2] = bv.z; db[3] = bv.w;
    }
    __syncthreads();

    for (int kk = 0; kk < 16; kk += 4) {
      v2f af[4], bf[2];
      for (int i = 0; i < 4; ++i) {
        const float* p = &As[(waveM + 16 * i + ml) * LPITCH + kk + kp];
        af[i].x = p[0]; af[i].y = p[1];
      }
      for (int j = 0; j < 2; ++j) {
        const float* p = &Bs[(waveN + 16 * j + ml) * LPITCH + kk + kp];
        bf[j].x = p[0]; bf[j].y = p[1];
      }
      for (int i = 0; i < 4; ++i)
        for (int j = 0; j < 2; ++j)
          acc[i][j] = __builtin_amdgcn_wmma_f32_16x16x4_f32(
              false, af[i], false, bf[j], (short)0, acc[i][j], false, false);
    }
    __syncthreads();
  }

  const float scale = scale_ptr ? scale_ptr[0] : 1.0f;
  const int rhalf = (lane < 16) ? 0 : 8;
  for (int i = 0; i < 4; ++i) {
    for (int j = 0; j < 2; ++j) {
      const int col  = bn + waveN + 16 * j + ml;
      const float bv = bias ? bias[col] : 0.0f;
      for (int r = 0; r < 8; ++r) {
        const int row = bm + waveM + 16 * i + r + rhalf;
        C[(size_t)row * ldc + colOff + col] = acc[i][j][r] * scale + bv;
      }
    }
  }
}

// ---------------------------------------------------------------------------
// Elementwise parameter epilogue, in-place on P (BT x 6N):
//  cols [0,N)->a, [N,2N)->b*u_s, [2N,3N)->c, [3N,4N)->d*u_s
// ---------------------------------------------------------------------------
__device__ __forceinline__ float softplusf(float x) {
  return fmaxf(x, 0.0f) + log1pf(expf(-fabsf(x)));
}

__global__ __launch_bounds__(256)
void param_epilogue_kernel(float* __restrict__ P,
                           const float* __restrict__ b_in,
                           const float* __restrict__ b_p,
                           const float* __restrict__ alpha_log,
                           const float* __restrict__ delta_bias,
                           const float* __restrict__ sign_bias,
                           const float* __restrict__ log_gamma,
                           const float* __restrict__ scal) {
  const long long idx = (long long)blockIdx.x * blockDim.x + threadIdx.x;
  if (idx >= (long long)BT * NN) return;
  const int n = (int)(idx & (NN - 1));
  const long long bt = idx >> 10;
  const size_t base = (size_t)bt * SIXN;

  const float inv_in = scal[0];
  const float gamma  = expf(log_gamma[0]);

  const float ut = P[base + n] * inv_in + b_in[n];
  const float us = gamma * ut;
  const float dr = P[base + 1 * NN + n] + b_p[0 * NN + n] + delta_bias[n];
  const float sr = P[base + 2 * NN + n] + b_p[1 * NN + n] + sign_bias[n];
  const float br = P[base + 3 * NN + n] + b_p[2 * NN + n];
  const float cr = P[base + 4 * NN + n] + b_p[3 * NN + n];
  const float er = P[base + 5 * NN + n] + b_p[4 * NN + n];

  const float alpha = softplusf(alpha_log[n]);
  const float delta = softplusf(dr);
  const float rho   = fminf(expf(-delta * alpha), 1.0f - 1e-4f);
  float a = rho * tanhf(sr);
  float b = tanhf(br);
  float c = tanhf(cr);
  float d = tanhf(er);

  // exact 2x2 spectral norm
  const float tau   = a * a + b * b + c * c + d * d;
  const float det   = a * d - b * c;
  float inner = tau * tau - 4.0f * det * det;
  inner = fmaxf(inner, 0.0f);
  const float lam   = 0.5f * (tau + sqrtf(inner + 1e-12f));
  const float sigma = sqrtf(lam + 1e-12f);
  const float inv   = 1.0f / fmaxf(sigma, 1.0f);
  a *= inv; b *= inv; c *= inv; d *= inv;

  P[base + 0 * NN + n] = a;
  P[base + 1 * NN + n] = b * us;   // bu
  P[base + 2 * NN + n] = c;
  P[base + 3 * NN + n] = d * us;   // y_pre
}

// ---------------------------------------------------------------------------
// Chunked scan: z_t = a_t z_{t-1} + bu_t ; y_t = c_t z_{t-1} + ypre_t
// ---------------------------------------------------------------------------
__global__ __launch_bounds__(256)
void scan_chunk_reduce_kernel(const float* __restrict__ P,
                              float* __restrict__ SA, float* __restrict__ SZ) {
  const int idx = blockIdx.x * blockDim.x + threadIdx.x;   // B*NC*N = 65536
  const int n = idx & (NN - 1);
  const int r = idx >> 10;
  const int j = r & (NC - 1);
  const int b = r >> 4;
  size_t base = ((size_t)b * TT + (size_t)j * CL) * SIXN + n;
  float Ap = 1.0f, Z = 0.0f;
  for (int t = 0; t < CL; ++t) {
    const float av  = P[base];
    const float buv = P[base + NN];
    Z  = av * Z + buv;
    Ap *= av;
    base += SIXN;
  }
  SA[idx] = Ap;
  SZ[idx] = Z;
}

__global__ __launch_bounds__(256)
void scan_chunk_combine_kernel(const float* __restrict__ SA,
                               const float* __restrict__ SZ,
                               float* __restrict__ ZS) {
  const int idx = blockIdx.x * blockDim.x + threadIdx.x;   // B*N = 4096
  const int n = idx & (NN - 1);
  const int b = idx >> 10;
  float z = 0.0f;   // z0 = 0
  for (int j = 0; j < NC; ++j) {
    const int s = (((b * NC) + j) << 10) + n;
    ZS[s] = z;
    z = SA[s] * z + SZ[s];
  }
}

__global__ __launch_bounds__(256)
void scan_replay_kernel(const float* __restrict__ P,
                        const float* __restrict__ ZS,
                        float* __restrict__ YH) {
  const int idx = blockIdx.x * blockDim.x + threadIdx.x;   // 65536
  const int n = idx & (NN - 1);
  const int r = idx >> 10;
  const int j = r & (NC - 1);
  const int b = r >> 4;
  float z = ZS[idx];
  size_t base  = ((size_t)b * TT + (size_t)j * CL) * SIXN + n;
  size_t ybase = ((size_t)b * TT + (size_t)j * CL) * NN + n;
  for (int t = 0; t < CL; ++t) {
    const float av   = P[base];
    const float buv  = P[base + 1 * NN];
    const float cv   = P[base + 2 * NN];
    const float ypre = P[base + 3 * NN];
    YH[ybase] = cv * z + ypre;     // output uses pre-update state
    z = av * z + buv;
    base  += SIXN;
    ybase += NN;
  }
}

// ---------------------------------------------------------------------------
extern "C" void kernel_launch(void* const* d_in, const int* in_sizes, int n_in,
                              void* d_out, int out_size, void* d_ws, size_t ws_size,
                              hipStream_t stream) {
  (void)in_sizes; (void)n_in; (void)out_size;
  const float* u          = (const float*)d_in[0];
  const float* log_gamma  = (const float*)d_in[1];
  const float* W_in       = (const float*)d_in[2];
  const float* b_in       = (const float*)d_in[3];
  const float* W_out      = (const float*)d_in[4];
  const float* b_out      = (const float*)d_in[5];
  const float* W_p        = (const float*)d_in[6];
  const float* b_p        = (const float*)d_in[7];
  const float* alpha_log  = (const float*)d_in[8];
  const float* delta_bias = (const float*)d_in[9];
  const float* sign_bias  = (const float*)d_in[10];
  float* out = (float*)d_out;
  float* ws  = (float*)d_ws;

  // workspace layout (floats)
  const size_t P_SZ  = (size_t)BT * SIXN;        // 50,331,648
  const size_t YH_SZ = (size_t)BT * NN;          //  8,388,608
  const size_t CH_SZ = (size_t)BB * NC * NN;     //     65,536
  float* P  = ws;
  float* YH = P  + P_SZ;
  float* SA = YH + YH_SZ;
  float* SZ = SA + CH_SZ;
  float* ZS = SZ + CH_SZ;
  float* SC = ZS + CH_SZ;                        // SC[0]=inv_in, SC[1]=inv_out
  const size_t need = (P_SZ + YH_SZ + 3 * CH_SZ + 16) * sizeof(float);
  if (ws_size < need) return;

  // 1) spectral-norm scales
  spectral_inv_scale_kernel<<<1, 1024, 0, stream>>>(W_in,  NN, SC + 0);
  spectral_inv_scale_kernel<<<1, 1024, 0, stream>>>(W_out, NN, SC + 1);

  // 2) P = u @ [W_in ; W_p]^T   (unscaled; scale folded into epilogue)
  gemm_wmma_f32_kernel<<<dim3(NN / 128, BT / 128), 256, 0, stream>>>(
      u, W_in, P, BT, NN, DD, SIXN, 0, nullptr, nullptr);
  gemm_wmma_f32_kernel<<<dim3((5 * NN) / 128, BT / 128), 256, 0, stream>>>(
      u, W_p, P, BT, 5 * NN, DD, SIXN, NN, nullptr, nullptr);

  // 3) elementwise params -> a, bu, c, ypre (in place)
  param_epilogue_kernel<<<(BT * NN) / 256, 256, 0, stream>>>(
      P, b_in, b_p, alpha_log, delta_bias, sign_bias, log_gamma, SC);

  // 4) chunked diagonal scan
  scan_chunk_reduce_kernel<<<(BB * NC * NN) / 256, 256, 0, stream>>>(P, SA, SZ);
  scan_chunk_combine_kernel<<<(BB * NN) / 256, 256, 0, stream>>>(SA, SZ, ZS);
  scan_replay_kernel<<<(BB * NC * NN) / 256, 256, 0, stream>>>(P, ZS, YH);

  // 5) y = (YH @ W_out^T) * inv_out + b_out
  gemm_wmma_f32_kernel<<<dim3(NN / 128, BT / 128), 256, 0, stream>>>(
      YH, W_out, out, BT, NN, NN, NN, 0, SC + 1, b_out);
}